// OrbitalCofactorAntiequivarianceLayer_12790412607704
// MI455X (gfx1250) — compile-verified
//
#include <hip/hip_runtime.h>
#include <math.h>

// ---------------------------------------------------------------------------
// Fused kernel: per-batch projection GEMM (WMMA f32 16x16x4) -> 32x32 LU with
// partial pivoting in LDS -> first-row-of-inverse solves -> slog cofactors ->
// streaming elementwise sign/log outputs.
// ---------------------------------------------------------------------------

typedef __attribute__((ext_vector_type(2))) float v2f;
typedef __attribute__((ext_vector_type(8))) float v8f;

#define BATCH 2048
#define NELEC 64
#define DIMK  256
#define BPB   4            // batches per block
#define MPB   (BPB * 2)    // matrices per block (up + down halves) = 8 waves
#define LP    33           // LDS row pitch (33 -> conflict free for 32 banks)
#define MATF  (32 * LP)    // floats per LDS matrix

__device__ __forceinline__ void wave_lds_fence() {
  // CDNA5 split dependency counter: wait for all outstanding LDS ops of this
  // wave, and act as a compiler memory barrier so LDS accesses are not
  // reordered across cross-lane dependences.
  asm volatile("s_wait_dscnt 0" ::: "memory");
}

__device__ __forceinline__ float sgnf(float x) {
  return (x > 0.f) ? 1.f : ((x < 0.f) ? -1.f : 0.f);
}

__global__ __launch_bounds__(256)
void cofactor_fused_kernel(const float* __restrict__ eq,
                           const float* __restrict__ Wup,
                           const float* __restrict__ Wdn,
                           float* __restrict__ outS,
                           float* __restrict__ outL)
{
  __shared__ float lds[MPB * MATF];
  __shared__ float cofS[MPB * 32];
  __shared__ float cofL[MPB * 32];

  const int tid  = threadIdx.x;
  const int wid  = tid >> 5;     // wave in block: owns one 32x32 matrix
  const int lane = tid & 31;
  const int hi   = lane >> 4;    // lane half (K-split for WMMA fragments)
  const int lr   = lane & 15;

  const int blk = blockIdx.x;
  const int bat = blk * BPB + (wid >> 1);
  const int h   = wid & 1;       // 0 = up-spin matrix, 1 = down-spin matrix

  float* A = lds + wid * MATF;

  // ------------------------------------------------------------------
  // Phase 1: M(32x32) = X(32x256) @ W(256x32) via V_WMMA_F32_16X16X4_F32
  // ------------------------------------------------------------------
  const float* W  = h ? Wdn : Wup;                                   // 256x32 row-major
  const float* x0 = eq + ((size_t)bat * NELEC + h * 32 + lr) * DIMK; // rows 0..15
  const float* x1 = x0 + 16 * DIMK;                                  // rows 16..31

  v8f acc00 = {}; v8f acc01 = {}; v8f acc10 = {}; v8f acc11 = {};

  for (int k = 0; k < DIMK; k += 4) {
    const int kk = k + 2 * hi;
    // A fragment (16x4 f32): lanes 0-15 hold K=k,k+1; lanes 16-31 hold K=k+2,k+3
    v2f a0 = *(const v2f*)(x0 + kk);
    v2f a1 = *(const v2f*)(x1 + kk);
    // B fragment (4x16 f32), mirrored K-split across lane halves
    v2f b0, b1;
    b0.x = W[kk * 32 + lr];            b0.y = W[(kk + 1) * 32 + lr];
    b1.x = W[kk * 32 + 16 + lr];       b1.y = W[(kk + 1) * 32 + 16 + lr];

    acc00 = __builtin_amdgcn_wmma_f32_16x16x4_f32(false, a0, false, b0, (short)0, acc00, false, false);
    acc01 = __builtin_amdgcn_wmma_f32_16x16x4_f32(false, a0, false, b1, (short)0, acc01, false, false);
    acc10 = __builtin_amdgcn_wmma_f32_16x16x4_f32(false, a1, false, b0, (short)0, acc10, false, false);
    acc11 = __builtin_amdgcn_wmma_f32_16x16x4_f32(false, a1, false, b1, (short)0, acc11, false, false);
  }

  // Scatter D tiles to padded LDS matrix. C/D layout: VGPR r, lanes 0-15 row r,
  // lanes 16-31 row r+8, column = lane%16.
#pragma unroll
  for (int r = 0; r < 8; ++r) {
    const int m = r + 8 * hi;
    A[m * LP + lr]             = acc00[r];
    A[m * LP + 16 + lr]        = acc01[r];
    A[(16 + m) * LP + lr]      = acc10[r];
    A[(16 + m) * LP + 16 + lr] = acc11[r];
  }
  wave_lds_fence();

  // ------------------------------------------------------------------
  // Phase 2: LU with partial pivoting (wave-cooperative, in LDS)
  // ------------------------------------------------------------------
  const float c0 = A[lane * LP + 0];   // original first column M[lane][0]
  int   perm   = lane;                 // original row now at position `lane`
  float parity = 1.f;

  for (int k = 0; k < 32; ++k) {
    // pivot search on column k, rows >= k
    float v  = (lane >= k) ? fabsf(A[lane * LP + k]) : -1.f;
    int  idx = lane;
#pragma unroll
    for (int off = 16; off > 0; off >>= 1) {
      float ov = __shfl_xor(v, off, 32);
      int   oi = __shfl_xor(idx, off, 32);
      if (ov > v || (ov == v && oi < idx)) { v = ov; idx = oi; }
    }
    const int piv = idx;               // uniform across wave
    if (piv != k) {
      parity = -parity;
      const int src = (lane == k) ? piv : ((lane == piv) ? k : lane);
      perm = __shfl(perm, src, 32);
      float tk = A[k * LP + lane];
      float tp = A[piv * LP + lane];
      wave_lds_fence();
      A[k * LP + lane]   = tp;
      A[piv * LP + lane] = tk;
      wave_lds_fence();
    }

    const float urow   = A[k * LP + lane];        // U[k][lane]
    const float pivval = __shfl(urow, k, 32);     // U[k][k]
    const float aik    = A[lane * LP + k];
    const float mult   = aik / pivval;
    if (lane > k) A[lane * LP + k] = mult;        // store L multiplier
    wave_lds_fence();

    if (lane > k) {
      for (int j = k + 1; j < 32; ++j) {
        const float ukj = __shfl(urow, j, 32);    // src lanes j>k are active
        A[lane * LP + j] -= mult * ukj;
      }
    }
    wave_lds_fence();
  }

  // det(M) = parity * prod(U[k][k]) in slog form
  const float diag = A[lane * LP + lane];
  float ld = logf(fabsf(diag));
  float sg = (diag < 0.f) ? -1.f : 1.f;
#pragma unroll
  for (int off = 16; off > 0; off >>= 1) {
    ld += __shfl_xor(ld, off, 32);
    sg *= __shfl_xor(sg, off, 32);
  }
  const float detSign   = parity * sg;
  const float logDetAbs = ld;

  // ------------------------------------------------------------------
  // Phase 3: row 0 of M^{-1}: solve M^T z = e0 via U^T w = e0, L^T y = w,
  // then z[perm[i]] = y[i].
  // ------------------------------------------------------------------
  float s = (lane == 0) ? 1.f : 0.f;
  float w = 0.f;
  for (int j = 0; j < 32; ++j) {
    const float wj = __shfl(s, j, 32) / __shfl(diag, j, 32);
    if (lane == j) w = wj;
    const float ujl = A[j * LP + lane];           // U[j][lane] for lane>j
    if (lane > j) s -= ujl * wj;
  }
  float t = w, y = 0.f;
  for (int j = 31; j >= 0; --j) {
    const float yj = __shfl(t, j, 32);            // unit diagonal
    if (lane == j) y = yj;
    const float ljl = A[j * LP + lane];           // L[j][lane] for lane<j
    if (lane < j) t -= ljl * yj;
  }

  // cofactor_i = M[r][0] * det(M) * z_r   (r = perm, z_r = y of this lane)
  const float m_r0 = __shfl(c0, perm, 32);
  const float scof = sgnf(m_r0) * detSign * sgnf(y);
  const float lcof = logf(fabsf(m_r0)) + logDetAbs + logf(fabsf(y));
  cofS[wid * 32 + perm] = scof;
  cofL[wid * 32 + perm] = lcof;
  __syncthreads();

  // ------------------------------------------------------------------
  // Phase 4: streaming elementwise outputs (bandwidth-dominant, float4)
  // sign_out = sign(eq) * sign_cof[b,n] ; log_out = log|eq| + log_cof[b,n]
  // ------------------------------------------------------------------
  const float4* eq4 = (const float4*)eq;
  float4* oS4 = (float4*)outS;
  float4* oL4 = (float4*)outL;
  const int    per_blk4 = BPB * NELEC * (DIMK / 4);       // 16384 float4
  const size_t base4    = (size_t)blk * per_blk4;

  for (int i = tid; i < per_blk4; i += 256) {
    const int nall = i >> 6;           // b_local*64 + n  (64 float4 per row)
    const int n    = nall & 63;
    const int bl   = nall >> 6;
    const int lw   = bl * 2 + (n >> 5);
    const int row  = n & 31;
    const float sc = cofS[lw * 32 + row];
    const float lc = cofL[lw * 32 + row];

    const float4 e = eq4[base4 + i];
    float4 so, lo;
    so.x = sgnf(e.x) * sc;  lo.x = logf(fabsf(e.x)) + lc;
    so.y = sgnf(e.y) * sc;  lo.y = logf(fabsf(e.y)) + lc;
    so.z = sgnf(e.z) * sc;  lo.z = logf(fabsf(e.z)) + lc;
    so.w = sgnf(e.w) * sc;  lo.w = logf(fabsf(e.w)) + lc;
    oS4[base4 + i] = so;
    oL4[base4 + i] = lo;
  }
}

extern "C" void kernel_launch(void* const* d_in, const int* in_sizes, int n_in,
                              void* d_out, int out_size, void* d_ws, size_t ws_size,
                              hipStream_t stream) {
  (void)in_sizes; (void)n_in; (void)out_size; (void)d_ws; (void)ws_size;
  const float* eq  = (const float*)d_in[0];
  const float* Wup = (const float*)d_in[1];
  const float* Wdn = (const float*)d_in[2];
  float* outS = (float*)d_out;
  float* outL = outS + (size_t)BATCH * NELEC * DIMK;

  dim3 grid(BATCH / BPB);
  dim3 block(256);
  hipLaunchKernelGGL(cofactor_fused_kernel, grid, block, 0, stream,
                     eq, Wup, Wdn, outS, outL);
}